// ModelNew_51659866636817
// MI455X (gfx1250) — compile-verified
//
#include <hip/hip_runtime.h>

typedef __attribute__((ext_vector_type(2))) float v2f;
typedef __attribute__((ext_vector_type(8))) float v8f;

#define S_LEN   2048
#define DH      64
#define QW      128          // query rows per workgroup (8 waves x 16)
#define KB      32           // keys per LDS block (2 tiles of 16)
#define KSTR    68           // K LDS row stride (dwords): 4*row mod 64 distinct -> conflict-free b64 reads
#define VSTR    72           // V LDS row stride (dwords): +16 bank shift between lane halves
#define EPS_N   1e-6f

__device__ __forceinline__ float swap16(float x) {
    // ds_swizzle SWAPX16: offset[14:10]=xor=0x10, [9:5]=or=0, [4:0]=and=0x1f
    int i = __builtin_bit_cast(int, x);
    i = __builtin_amdgcn_ds_swizzle(i, 0x401F);
    return __builtin_bit_cast(float, i);
}

__global__ __launch_bounds__(256) void taylor_attn_fwd(
    const float* __restrict__ q,
    const float* __restrict__ k,
    const float* __restrict__ v,
    float* __restrict__ out)
{
    __shared__ float kbuf[KB * KSTR];
    __shared__ float vbuf[KB * VSTR];

    const int tid  = threadIdx.x;
    const int lane = tid & 31;
    // wave index is uniform across the wave: pin it to an SGPR so the causal
    // tile filter becomes a scalar branch (s_cmp/s_cbranch), not v_cmpx/EXEC.
    const int wave = __builtin_amdgcn_readfirstlane(tid >> 5);
    const int lrow = lane & 15;   // query/key row within a 16-tile
    const int lhi  = lane >> 4;   // lane half

    const int qblk = blockIdx.x & 15;        // S/QW = 16 query blocks per head
    const int bh   = blockIdx.x >> 4;        // 0..31 (b*h)
    const long long base = (long long)bh * S_LEN * DH;

    const int qbase_w = qblk * QW + wave * 16;   // this wave's 16 query rows (SGPR)
    const float scale = 0.125f;                  // 64^-0.5

    // ---- Q tile, prescaled, in B-operand layout: lane holds row (lane&15),
    //      d-pair (4j + 2*lhi + {0,1}) for j = 0..15 ----
    v2f qreg[16];
    {
        const float* qp = q + base + (long long)(qbase_w + lrow) * DH + 2 * lhi;
        #pragma unroll
        for (int j = 0; j < 16; ++j) {
            v2f t = *(const v2f*)(qp + 4 * j);
            qreg[j] = t * scale;
        }
    }

    v8f accO[4] = {};   // O^T accumulators, 4 tiles of 16 d-columns
    v8f accZ   = {};    // normalizer (all M-rows identical)
    v2f ones;  ones.x = 1.0f; ones.y = 1.0f;

    const int nblocks = (qblk + 1) * (QW / KB);  // WG-wide causal key range

    // ---- software-pipelined staging: regs hold block blk's K/V before commit ----
    const int r = tid >> 3;            // 0..31
    const int c = (tid & 7) * 8;       // 0,8,...,56
    const float* kp0 = k + base + (long long)r * DH + c;
    const float* vp0 = v + base + (long long)r * DH + c;

    float4 rk0 = *(const float4*)(kp0);
    float4 rk1 = *(const float4*)(kp0 + 4);
    float4 rv0 = *(const float4*)(vp0);
    float4 rv1 = *(const float4*)(vp0 + 4);

    for (int blk = 0; blk < nblocks; ++blk) {
        // commit staged block to LDS
        *(float4*)(&kbuf[r * KSTR + c])     = rk0;
        *(float4*)(&kbuf[r * KSTR + c + 4]) = rk1;
        *(float4*)(&vbuf[r * VSTR + c])     = rv0;
        *(float4*)(&vbuf[r * VSTR + c + 4]) = rv1;
        __syncthreads();

        // issue next block's global loads now; their wait lands at the next
        // iteration's ds_store, overlapping latency with the WMMA body below
        if (blk + 1 < nblocks) {
            const float* kp = kp0 + (long long)(blk + 1) * KB * DH;
            const float* vp = vp0 + (long long)(blk + 1) * KB * DH;
            rk0 = *(const float4*)(kp);
            rk1 = *(const float4*)(kp + 4);
            rv0 = *(const float4*)(vp);
            rv1 = *(const float4*)(vp + 4);
        }

        const int kb_g0 = blk * KB;
        #pragma unroll
        for (int kt = 0; kt < 2; ++kt) {
            const int kb_g = kb_g0 + kt * 16;
            if (kb_g <= qbase_w) {   // scalar causal tile filter
                // ---- GEMM1: P[key][q] = K_tile x Qs^T, f32 WMMA chained over d ----
                const float* krow = &kbuf[(kt * 16 + lrow) * KSTR + 2 * lhi];
                v8f p = {};
                #pragma unroll
                for (int j = 0; j < 16; ++j) {
                    v2f a = *(const v2f*)(krow + 4 * j);
                    p = __builtin_amdgcn_wmma_f32_16x16x4_f32(
                            false, a, false, qreg[j], (short)0, p, false, false);
                }

                // ---- weight w = 1 + s + 0.5 s^2, causal mask on diagonal tile ----
                // C/D layout: (key = i + 8*lhi, q = lane&15)
                float w[8];
                const bool diag = (kb_g == qbase_w);
                #pragma unroll
                for (int i = 0; i < 8; ++i) {
                    float s  = p[i];
                    float ww = fmaf(0.5f * s, s, s) + 1.0f;
                    if (diag) {
                        int key_g = kb_g + i + 8 * lhi;
                        int q_g   = qbase_w + lrow;
                        ww = (key_g <= q_g) ? ww : 0.0f;
                    }
                    w[i] = ww;
                }

                // ---- C/D -> B-operand layout: fixed lane-half swap + select ----
                float sw[8];
                #pragma unroll
                for (int i = 0; i < 8; ++i) sw[i] = swap16(w[i]);
                const bool hi = (lane >= 16);
                v2f bw[4];
                bw[0].x = hi ? sw[2] : w[0];  bw[0].y = hi ? sw[3] : w[1];  // keys 0..3
                bw[1].x = hi ? sw[6] : w[4];  bw[1].y = hi ? sw[7] : w[5];  // keys 4..7
                bw[2].x = hi ? w[2] : sw[0];  bw[2].y = hi ? w[3] : sw[1];  // keys 8..11
                bw[3].x = hi ? w[6] : sw[4];  bw[3].y = hi ? w[7] : sw[5];  // keys 12..15

                // ---- GEMM2: O^T += V^T x W  (+ z += 1^T x W) ----
                #pragma unroll
                for (int kb = 0; kb < 4; ++kb) {
                    #pragma unroll
                    for (int ft = 0; ft < 4; ++ft) {
                        const int keyl = kt * 16 + kb * 4 + 2 * lhi;
                        v2f a;
                        a.x = vbuf[keyl * VSTR + ft * 16 + lrow];
                        a.y = vbuf[(keyl + 1) * VSTR + ft * 16 + lrow];
                        accO[ft] = __builtin_amdgcn_wmma_f32_16x16x4_f32(
                                false, a, false, bw[kb], (short)0, accO[ft], false, false);
                    }
                    accZ = __builtin_amdgcn_wmma_f32_16x16x4_f32(
                            false, ones, false, bw[kb], (short)0, accZ, false, false);
                }
            }
        }
        __syncthreads();
    }

    // ---- epilogue: divide by z (lane-local) and store as b128 ----
    // accO value (lane, i) -> out row q = lane&15, col f = 16*ft + i + 8*lhi
    const float inv = 1.0f / (accZ[0] + EPS_N);
    float* op = out + base + (long long)(qbase_w + lrow) * DH + 8 * lhi;
    #pragma unroll
    for (int ft = 0; ft < 4; ++ft) {
        float4 lo, hiv;
        lo.x  = accO[ft][0] * inv;  lo.y  = accO[ft][1] * inv;
        lo.z  = accO[ft][2] * inv;  lo.w  = accO[ft][3] * inv;
        hiv.x = accO[ft][4] * inv;  hiv.y = accO[ft][5] * inv;
        hiv.z = accO[ft][6] * inv;  hiv.w = accO[ft][7] * inv;
        *(float4*)(op + ft * 16)     = lo;   // 16B aligned: 8*lhi + 16*ft offsets
        *(float4*)(op + ft * 16 + 4) = hiv;
    }
}

extern "C" void kernel_launch(void* const* d_in, const int* in_sizes, int n_in,
                              void* d_out, int out_size, void* d_ws, size_t ws_size,
                              hipStream_t stream) {
    (void)n_in; (void)out_size; (void)d_ws; (void)ws_size;
    const float* q = (const float*)d_in[0];
    const float* k = (const float*)d_in[1];
    const float* v = (const float*)d_in[2];
    float* out     = (float*)d_out;

    const int bh = in_sizes[0] / (S_LEN * DH);      // b*h = 32
    dim3 grid(bh * (S_LEN / QW));                   // 512 workgroups
    taylor_attn_fwd<<<grid, 256, 0, stream>>>(q, k, v, out);
}